// AttnDecoderRNN_59081570124588
// MI455X (gfx1250) — compile-verified
//
#include <hip/hip_runtime.h>

#define H  2048
#define V  50257
#define ML 25

typedef __attribute__((ext_vector_type(2))) float v2f;
typedef __attribute__((ext_vector_type(8))) float v8f;

// ws layout (floats)
#define WS_EMB  0          // [H]   embedded
#define WS_ATT  (H)        // [H]   attn_applied   (WS_EMB..WS_ATT+H is concat(x))
#define WS_COMB (2*H)      // [H]   relu(combined)
#define WS_GI   (3*H)      // [3H]  gi
#define WS_GH   (6*H)      // [3H]  gh
// total = 9*H = 18432 floats = 73728 bytes

// ---------------------------------------------------------------------------
// Wave-level GEMV over 16 rows using V_WMMA_F32_16X16X4_F32.
// A (16x4 fp32) = W tile: lane m=lane&15 is the row, lane bit4 selects K pair,
// each lane loads float2 -> matches ISA 16x4 A layout exactly.
// B (4x16) = x chunk broadcast to all 16 columns (same float2 load from x),
// so every accumulator column holds the identical dot product:
// lane 0 ends with y[r0..r0+7] in c[0..7], lane 16 with y[r0+8..r0+15].
// All 32 lanes must be active (EXEC all ones) — callers guarantee this by
// clamping row indices instead of branching out.
// ---------------------------------------------------------------------------
__device__ __forceinline__ void wave_gemv16(const float* __restrict__ W, int ldw,
                                            const float* __restrict__ x, int K,
                                            int r0, int nrows,
                                            const float* __restrict__ bias,
                                            float* __restrict__ y, bool do_relu)
{
    const int lane  = threadIdx.x & 31;
    const int m     = lane & 15;
    const int khalf = lane >> 4;          // 0 or 1
    int row = r0 + m;
    if (row >= nrows) row = nrows - 1;    // clamp: keep loads valid, EXEC full

    const float* wrow = W + (size_t)row * (size_t)ldw + 2 * khalf;
    const float* xp   = x + 2 * khalf;

    v8f c = {0.f, 0.f, 0.f, 0.f, 0.f, 0.f, 0.f, 0.f};
#pragma unroll 4
    for (int k0 = 0; k0 < K; k0 += 4) {
        v2f a = *(const v2f*)(wrow + k0);
        v2f b = *(const v2f*)(xp + k0);
        c = __builtin_amdgcn_wmma_f32_16x16x4_f32(
                /*neg_a=*/false, a, /*neg_b=*/false, b,
                /*c_mod=*/(short)0, c, /*reuse_a=*/false, /*reuse_b=*/false);
    }

    // lane 0 writes rows r0..r0+7, lane 16 writes rows r0+8..r0+15
    if (lane == 0 || lane == 16) {
        const int base = r0 + (lane >> 1);
#pragma unroll
        for (int i = 0; i < 8; ++i) {
            const int r = base + i;
            if (r < nrows) {
                float v = c[i] + bias[r];
                y[r] = do_relu ? fmaxf(v, 0.f) : v;
            }
        }
    }
}

// ---------------------------------------------------------------------------
// K1: embedding gather, attention scores (25 x 4096 dots), softmax(25),
//     attn_applied = attn_w @ encoder_outputs. Single block, 256 threads.
// ---------------------------------------------------------------------------
__global__ void k1_attn(const int* __restrict__ tok_p,
                        const float* __restrict__ hidden,
                        const float* __restrict__ enc_outputs,
                        const float* __restrict__ emb,
                        const float* __restrict__ W_attn,
                        const float* __restrict__ b_attn,
                        float* __restrict__ out_attnw,
                        float* __restrict__ ws)
{
    __shared__ float sw[ML];
    const int tid  = threadIdx.x;
    const int wave = tid >> 5;
    const int lane = tid & 31;
    const int tok  = tok_p[0];   // int64 LE: low dword is the token value

    // embedded -> ws[WS_EMB..]
    for (int i = tid; i < H; i += 256) ws[WS_EMB + i] = emb[(size_t)tok * H + i];
    __syncthreads();

    // scores[j] = W_attn[j] . concat(embedded, h0) + b_attn[j]
    for (int j = wave; j < ML; j += 8) {
        const float* wr = W_attn + (size_t)j * (2 * H);
        float acc = 0.f;
        for (int i = lane; i < H; i += 32) acc += wr[i] * ws[WS_EMB + i];
        for (int i = lane; i < H; i += 32) acc += wr[H + i] * hidden[i];
        for (int off = 16; off > 0; off >>= 1) acc += __shfl_down(acc, off, 32);
        if (lane == 0) sw[j] = acc + b_attn[j];
    }
    __syncthreads();

    // softmax over 25 (serial in thread 0 — trivial)
    if (tid == 0) {
        float mx = sw[0];
        for (int j = 1; j < ML; ++j) mx = fmaxf(mx, sw[j]);
        float s = 0.f;
        for (int j = 0; j < ML; ++j) { sw[j] = expf(sw[j] - mx); s += sw[j]; }
        const float inv = 1.f / s;
        for (int j = 0; j < ML; ++j) sw[j] *= inv;
    }
    __syncthreads();

    if (tid < ML) out_attnw[tid] = sw[tid];

    // attn_applied[i] = sum_j sw[j] * enc_outputs[j][i]
    for (int i = tid; i < H; i += 256) {
        float acc = 0.f;
#pragma unroll
        for (int j = 0; j < ML; ++j) acc += sw[j] * enc_outputs[(size_t)j * H + i];
        ws[WS_ATT + i] = acc;
    }
}

// K2: combined = relu(W_comb @ concat(emb, attn) + b_comb)   [H rows, K=2H]
__global__ void k2_comb(const float* __restrict__ W_comb,
                        const float* __restrict__ b_comb,
                        float* __restrict__ ws)
{
    const int gw = blockIdx.x * (blockDim.x >> 5) + (threadIdx.x >> 5);
    wave_gemv16(W_comb, 2 * H, ws + WS_EMB, 2 * H, gw * 16, H,
                b_comb, ws + WS_COMB, /*relu=*/true);
}

// K3: gi = W_ih @ combined + b_ih ; gh = W_hh @ h0 + b_hh    [2 x 3H rows, K=H]
__global__ void k3_gru_mv(const float* __restrict__ W_ih,
                          const float* __restrict__ W_hh,
                          const float* __restrict__ b_ih,
                          const float* __restrict__ b_hh,
                          const float* __restrict__ hidden,
                          float* __restrict__ ws)
{
    const int gw = blockIdx.x * (blockDim.x >> 5) + (threadIdx.x >> 5);
    const int r0 = gw * 16;
    if (r0 < 3 * H) {   // wave-uniform branch (3H divisible by 16)
        wave_gemv16(W_ih, H, ws + WS_COMB, H, r0, 3 * H, b_ih, ws + WS_GI, false);
    } else {
        wave_gemv16(W_hh, H, hidden, H, r0 - 3 * H, 3 * H, b_hh, ws + WS_GH, false);
    }
}

// K4: GRU gating -> h_new into d_out[V .. V+H)
__global__ void k4_gate(const float* __restrict__ ws,
                        const float* __restrict__ hidden,
                        float* __restrict__ hnew)
{
    const int i = blockIdx.x * blockDim.x + threadIdx.x;
    if (i >= H) return;
    const float* gi = ws + WS_GI;
    const float* gh = ws + WS_GH;
    const float r = 1.f / (1.f + expf(-(gi[i] + gh[i])));
    const float z = 1.f / (1.f + expf(-(gi[H + i] + gh[H + i])));
    const float n = tanhf(gi[2 * H + i] + r * gh[2 * H + i]);
    hnew[i] = (1.f - z) * n + z * hidden[i];
}

// K5: logits = W_out @ h_new + b_out  -> d_out[0..V)   [V rows, K=H]
__global__ void k5_out(const float* __restrict__ W_out,
                       const float* __restrict__ b_out,
                       const float* __restrict__ hnew,
                       float* __restrict__ logits)
{
    const int gw = blockIdx.x * (blockDim.x >> 5) + (threadIdx.x >> 5);
    wave_gemv16(W_out, H, hnew, H, gw * 16, V, b_out, logits, false);
}

// K6: in-place log_softmax over d_out[0..V). Single 1024-thread block.
__global__ void k6_logsoftmax(float* __restrict__ logits)
{
    __shared__ float red[1024];
    const int tid = threadIdx.x;

    float m = -3.402823466e+38f;
    for (int i = tid; i < V; i += 1024) m = fmaxf(m, logits[i]);
    red[tid] = m; __syncthreads();
    for (int s = 512; s > 0; s >>= 1) {
        if (tid < s) red[tid] = fmaxf(red[tid], red[tid + s]);
        __syncthreads();
    }
    const float mx = red[0];
    __syncthreads();

    float acc = 0.f;
    for (int i = tid; i < V; i += 1024) acc += expf(logits[i] - mx);
    red[tid] = acc; __syncthreads();
    for (int s = 512; s > 0; s >>= 1) {
        if (tid < s) red[tid] += red[tid + s];
        __syncthreads();
    }
    const float lse = mx + logf(red[0]);

    for (int i = tid; i < V; i += 1024) logits[i] -= lse;
}

extern "C" void kernel_launch(void* const* d_in, const int* in_sizes, int n_in,
                              void* d_out, int out_size, void* d_ws, size_t ws_size,
                              hipStream_t stream)
{
    const int*   tok      = (const int*)  d_in[0];   // int64 scalar; low dword
    const float* hidden   = (const float*)d_in[1];   // [1,1,H]
    // d_in[2] = encoder_output (unused by the reference)
    const float* enc_outs = (const float*)d_in[3];   // [ML,H]
    const float* emb      = (const float*)d_in[4];   // [V,H]
    const float* W_attn   = (const float*)d_in[5];   // [ML,2H]
    const float* b_attn   = (const float*)d_in[6];   // [ML]
    const float* W_comb   = (const float*)d_in[7];   // [H,2H]
    const float* b_comb   = (const float*)d_in[8];   // [H]
    const float* W_ih     = (const float*)d_in[9];   // [3H,H]
    const float* W_hh     = (const float*)d_in[10];  // [3H,H]
    const float* b_ih     = (const float*)d_in[11];  // [3H]
    const float* b_hh     = (const float*)d_in[12];  // [3H]
    const float* W_out    = (const float*)d_in[13];  // [V,H]
    const float* b_out    = (const float*)d_in[14];  // [V]

    float* out = (float*)d_out;            // [V] log_probs | [H] h_new | [ML] attn_w
    float* ws  = (float*)d_ws;             // 9*H floats used

    k1_attn<<<1, 256, 0, stream>>>(tok, hidden, enc_outs, emb, W_attn, b_attn,
                                   out + V + H, ws);

    // H rows / 16 per wave = 128 waves = 16 blocks of 8 waves
    k2_comb<<<16, 256, 0, stream>>>(W_comb, b_comb, ws);

    // 2*3H rows -> 768 waves -> 96 blocks
    k3_gru_mv<<<96, 256, 0, stream>>>(W_ih, W_hh, b_ih, b_hh, hidden, ws);

    k4_gate<<<(H + 255) / 256, 256, 0, stream>>>(ws, hidden, out + V);

    // ceil(V/16) = 3142 waves -> 393 blocks (tail rows clamped inside)
    k5_out<<<393, 256, 0, stream>>>(W_out, b_out, out + V, out);

    k6_logsoftmax<<<1, 1024, 0, stream>>>(out);
}